// DocRE_75376676044824
// MI455X (gfx1250) — compile-verified
//
#include <hip/hip_runtime.h>
#include <hip/hip_bf16.h>

// ---------------------------------------------------------------------------
// DocRE pair-scoring pipeline for MI455X (gfx1250, wave32).
// Matrix ops: V_WMMA_F32_16X16X4_F32. LDS staging: GLOBAL_LOAD_ASYNC_TO_LDS_B128
// (ASYNCcnt) with double-buffered tiles so copy(k+1) overlaps compute(k).
// Shapes: B=4, S=1024, H=768, Hh=12, N=32, M=4, Wd=768, d=12, SPLIT_K=64,
//         AD=256, C=97
// ---------------------------------------------------------------------------

#define Bq   4
#define Sq   1024
#define Hq   768
#define HHq  12
#define Nq   32
#define Mq   4
#define WDq  768
#define DDq  12
#define SKq  64
#define ADq  256
#define Cq   97

typedef float v2f __attribute__((ext_vector_type(2)));
typedef float v8f __attribute__((ext_vector_type(8)));
typedef int   v4i __attribute__((ext_vector_type(4)));

typedef __attribute__((address_space(1))) v4i as1_v4i;   // global 16B chunk
typedef __attribute__((address_space(3))) v4i as3_v4i;   // LDS 16B chunk

// 16-byte global -> LDS async copy (per-lane addresses, tracked by ASYNCcnt)
__device__ __forceinline__ void asyncCopy16(const float* g, float* l)
{
#if defined(__HIP_DEVICE_COMPILE__)
#if __has_builtin(__builtin_amdgcn_global_load_async_to_lds_b128)
    __builtin_amdgcn_global_load_async_to_lds_b128(
        (as1_v4i*)g, (as3_v4i*)l, 0, 0);
#else
    unsigned ldsOff = (unsigned)(unsigned long long)(as3_v4i*)l;
    asm volatile("global_load_async_to_lds_b128 %0, %1, off"
                 :: "v"(ldsOff), "v"(g) : "memory");
#endif
#endif
}

__device__ __forceinline__ void waitAsync0()
{
#if defined(__HIP_DEVICE_COMPILE__)
#if __has_builtin(__builtin_amdgcn_s_wait_asynccnt)
    __builtin_amdgcn_s_wait_asynccnt(0);
#else
    asm volatile("s_wait_asynccnt 0" ::: "memory");
#endif
#endif
}

// ---------------------------------------------------------------------------
// Generic f32 WMMA GEMM: C = op(A @ B + biasScale*bias), row-major, batched.
// Block: 256 threads = 8 waves. Tile BM x BN = 64x128, BK = 32.
// Waves 4x2; each wave computes 16x64 (4 accumulators) -> 4x A-frag reuse.
// Double-buffered LDS, async global->LDS staging on full aligned tiles.
// ---------------------------------------------------------------------------
#define BM 64
#define BN 128
#define BK 32
#define PAD 4   // 16B pad: keeps B128 LDS writes aligned, reads conflict-free

__global__ __launch_bounds__(256)
void gemm_f32_wmma(const float* __restrict__ A, const float* __restrict__ Bm,
                   float* __restrict__ C,
                   int M, int N, int K, int lda, int ldb, int ldc,
                   long long strideA, long long strideB, long long strideC,
                   const float* __restrict__ bias, float biasScale, int doTanh)
{
    __shared__ float sA[2][BM][BK + PAD];
    __shared__ float sB[2][BK][BN + PAD];

    const int bz = blockIdx.z;
    A  += (long long)bz * strideA;
    Bm += (long long)bz * strideB;
    C  += (long long)bz * strideC;

    const int tid  = threadIdx.x;
    const int wave = tid >> 5;
    const int lane = tid & 31;
    const int half = lane >> 4;
    const int l16  = lane & 15;
    const int wr   = wave >> 1;     // 0..3
    const int wc   = wave & 1;      // 0..1

    const int mBase = blockIdx.y * BM;
    const int nBase = blockIdx.x * BN;
    const bool fullM   = (mBase + BM <= M);
    const bool fullN   = (nBase + BN <= N);
    const bool aligned = ((lda & 3) == 0) && ((ldb & 3) == 0);

    v8f acc[4];
#pragma unroll
    for (int q = 0; q < 4; ++q)
        acc[q] = (v8f){0.f,0.f,0.f,0.f,0.f,0.f,0.f,0.f};

    auto stage = [&](int buf, int k0) {
        if (fullM && fullN && aligned && (k0 + BK <= K)) {
            // Async path: each lane copies 16B straight into LDS.
#pragma unroll
            for (int r = 0; r < 2; ++r) {              // A: 512 chunks
                int ch = tid + r * 256;
                int row = ch >> 3, c4 = (ch & 7) << 2;
                asyncCopy16(A + (long long)(mBase + row) * lda + k0 + c4,
                            &sA[buf][row][c4]);
            }
#pragma unroll
            for (int r = 0; r < 4; ++r) {              // B: 1024 chunks
                int ch = tid + r * 256;
                int row = ch >> 5, c4 = (ch & 31) << 2;
                asyncCopy16(Bm + (long long)(k0 + row) * ldb + nBase + c4,
                            &sB[buf][row][c4]);
            }
        } else {
            // Guarded scalar path for edge tiles (zero-filled OOB).
            for (int t = tid; t < BM * BK; t += 256) {
                int r = t / BK, c = t % BK;
                int gr = mBase + r, gc = k0 + c;
                sA[buf][r][c] = (gr < M && gc < K) ? A[(long long)gr * lda + gc] : 0.f;
            }
            for (int t = tid; t < BK * BN; t += 256) {
                int r = t / BN, c = t % BN;
                int gr = k0 + r, gc = nBase + c;
                sB[buf][r][c] = (gr < K && gc < N) ? Bm[(long long)gr * ldb + gc] : 0.f;
            }
        }
    };

    stage(0, 0);
    int buf = 0;
    for (int k0 = 0; k0 < K; k0 += BK, buf ^= 1) {
        waitAsync0();        // own async copies of tile(k0) landed in LDS
        __syncthreads();     // all waves' copies visible
        if (k0 + BK < K) stage(buf ^ 1, k0 + BK);   // prefetch next tile

        const int aRow = wr * 16 + l16;
#pragma unroll
        for (int kk = 0; kk < BK; kk += 4) {
            // A 16x4: lanes 0-15 hold K=kk,kk+1; lanes 16-31 hold K=kk+2,kk+3
            v2f a;
            a.x = sA[buf][aRow][kk + 2 * half + 0];
            a.y = sA[buf][aRow][kk + 2 * half + 1];
#pragma unroll
            for (int q = 0; q < 4; ++q) {
                // B 4x16: n = lane%16; lower lanes K=kk,kk+1; upper K=kk+2,kk+3
                v2f b;
                b.x = sB[buf][kk + 2 * half + 0][wc * 64 + q * 16 + l16];
                b.y = sB[buf][kk + 2 * half + 1][wc * 64 + q * 16 + l16];
                acc[q] = __builtin_amdgcn_wmma_f32_16x16x4_f32(
                             false, a, false, b, (short)0, acc[q], false, false);
            }
        }
    }

    // D layout: m = v + 8*half, n = lane%16
#pragma unroll
    for (int v = 0; v < 8; ++v) {
        int m = mBase + wr * 16 + 8 * half + v;
        if (m >= M) continue;
#pragma unroll
        for (int q = 0; q < 4; ++q) {
            int n = nBase + wc * 64 + q * 16 + l16;
            if (n >= N) continue;
            float x = acc[q][v];
            if (bias) x += biasScale * bias[n];
            if (doTanh) x = tanhf(x);
            C[(long long)m * ldc + n] = x;
        }
    }
}

// ---------------------------------------------------------------------------
// Block-wide sum reduction helper (blockDim.x == 256)
// ---------------------------------------------------------------------------
__device__ __forceinline__ float blockReduceSum256(float v, float* sbuf)
{
    int tid = threadIdx.x;
    sbuf[tid] = v;
    __syncthreads();
    for (int s = 128; s > 0; s >>= 1) {
        if (tid < s) sbuf[tid] += sbuf[tid + s];
        __syncthreads();
    }
    float r = sbuf[0];
    __syncthreads();
    return r;
}

// ---------------------------------------------------------------------------
// rep[b,n,h] = logsumexp_m seq[b, ep[b,n,m], h]        grid(N,B), block 256
// ---------------------------------------------------------------------------
__global__ __launch_bounds__(256)
void k_rep(const float* __restrict__ seq, const int* __restrict__ ep,
           float* __restrict__ rep)
{
    int b = blockIdx.y, n = blockIdx.x;
    int pos[Mq];
#pragma unroll
    for (int m = 0; m < Mq; ++m) pos[m] = ep[(b * Nq + n) * Mq + m];

    for (int h = threadIdx.x; h < Hq; h += 256) {
        float x[Mq], mx = -1e30f;
#pragma unroll
        for (int m = 0; m < Mq; ++m) {
            x[m] = seq[((long long)b * Sq + pos[m]) * Hq + h];
            mx = fmaxf(mx, x[m]);
        }
        float s = 0.f;
#pragma unroll
        for (int m = 0; m < Mq; ++m) s += expf(x[m] - mx);
        rep[((long long)b * Nq + n) * Hq + h] = mx + logf(s);
    }
}

// ---------------------------------------------------------------------------
// ga[b,n,h,s] = sum_m att[b,h,ep[b,n,m],s]   grid(Hh,N,B), block 256
// Only 128 of 1024 attention rows per head are ever read (25MB vs 201MB).
// ---------------------------------------------------------------------------
__global__ __launch_bounds__(256)
void k_ga(const float* __restrict__ att, const int* __restrict__ ep,
          float* __restrict__ ga)
{
    int h = blockIdx.x, n = blockIdx.y, b = blockIdx.z;
    int pos[Mq];
#pragma unroll
    for (int m = 0; m < Mq; ++m) pos[m] = ep[(b * Nq + n) * Mq + m];

    const float* base = att + ((long long)(b * HHq + h)) * Sq * Sq;
    float* dst = ga + (((long long)(b * Nq + n)) * HHq + h) * Sq;
    for (int s = threadIdx.x; s < Sq; s += 256) {
        float v = 0.f;
#pragma unroll
        for (int m = 0; m < Mq; ++m) v += base[(long long)pos[m] * Sq + s];
        dst[s] = v;
    }
}

// ---------------------------------------------------------------------------
// pa[b,p,s] = (sum_h ga[i,h,s]*ga[j,h,s]) / (sum_s of same)   p=(i,j)
// grid(N*N, B), block 256; each thread owns 4 s-positions.
// ---------------------------------------------------------------------------
__global__ __launch_bounds__(256)
void k_pa(const float* __restrict__ ga, float* __restrict__ pa)
{
    __shared__ float sbuf[256];
    int b = blockIdx.y, p = blockIdx.x;
    int i = p >> 5, j = p & 31;
    const float* gi = ga + ((long long)(b * Nq + i)) * HHq * Sq;
    const float* gj = ga + ((long long)(b * Nq + j)) * HHq * Sq;

    float vals[4], part = 0.f;
#pragma unroll
    for (int r = 0; r < 4; ++r) {
        int s = threadIdx.x + r * 256;
        float v = 0.f;
#pragma unroll
        for (int h = 0; h < HHq; ++h) v += gi[h * Sq + s] * gj[h * Sq + s];
        vals[r] = v;
        part += v;
    }
    float tot = blockReduceSum256(part, sbuf);
    float inv = 1.f / tot;
    float* dst = pa + ((long long)b * (Nq * Nq) + p) * Sq;
#pragma unroll
    for (int r = 0; r < 4; ++r) dst[threadIdx.x + r * 256] = vals[r] * inv;
}

// ---------------------------------------------------------------------------
// Zs/Zo (tanh of rep-projection + ctx, both use row index i) and the per-pair
// grouped outer-product accumulation:
//   Pvec[p, t*12+u] = sum_{k<64} Zs[p, t*64+k] * Zo[p, u*64+k]
// grid(N*N, B), block 256.
// ---------------------------------------------------------------------------
__global__ __launch_bounds__(256)
void k_zsP(const float* __restrict__ ctx, const float* __restrict__ repWs,
           const float* __restrict__ repWo, float* __restrict__ Pvec)
{
    __shared__ float zs[WDq];
    __shared__ float zo[WDq];
    int b = blockIdx.y, p = blockIdx.x;
    int i = p >> 5;
    const float* c  = ctx   + ((long long)b * (Nq * Nq) + p) * WDq;
    const float* rs = repWs + ((long long)(b * Nq + i)) * WDq;
    const float* ro = repWo + ((long long)(b * Nq + i)) * WDq;

    for (int w = threadIdx.x; w < WDq; w += 256) {
        float cv = c[w];
        zs[w] = tanhf(rs[w] + cv);
        zo[w] = tanhf(ro[w] + cv);
    }
    __syncthreads();

    if (threadIdx.x < DDq * DDq) {
        int t = threadIdx.x / DDq, u = threadIdx.x % DDq;
        float s = 0.f;
#pragma unroll
        for (int k = 0; k < SKq; ++k) s += zs[t * SKq + k] * zo[u * SKq + k];
        Pvec[((long long)b * (Nq * Nq) + p) * (DDq * DDq) + threadIdx.x] = s;
    }
}

// ---------------------------------------------------------------------------
// AblT[k, o] = A_bl[o, k]   (144 x 768), coalesced writes
// ---------------------------------------------------------------------------
__global__ __launch_bounds__(256)
void k_transAbl(const float* __restrict__ Abl, float* __restrict__ AblT)
{
    int idx = blockIdx.x * 256 + threadIdx.x;
    if (idx >= DDq * DDq * WDq) return;
    int k = idx / WDq, o = idx % WDq;
    AblT[idx] = Abl[(long long)o * (DDq * DDq) + k];
}

// ---------------------------------------------------------------------------
// e[b,p] = sum_a tanhT[b,p,a] * v_attn[a]   (AD == 256 == blockDim)
// ---------------------------------------------------------------------------
__global__ __launch_bounds__(256)
void k_e(const float* __restrict__ tanhT, const float* __restrict__ vattn,
         float* __restrict__ e)
{
    __shared__ float sbuf[256];
    int b = blockIdx.y, p = blockIdx.x;
    float v = tanhT[((long long)b * (Nq * Nq) + p) * ADq + threadIdx.x] *
              vattn[threadIdx.x];
    float tot = blockReduceSum256(v, sbuf);
    if (threadIdx.x == 0) e[(long long)b * (Nq * Nq) + p] = tot;
}

// ---------------------------------------------------------------------------
// Row (axis=1) and column (axis=0) softmax of e[32,32]. grid(B), block 64.
// ---------------------------------------------------------------------------
__global__ __launch_bounds__(64)
void k_softmax_axes(const float* __restrict__ e, float* __restrict__ arow,
                    float* __restrict__ acol)
{
    __shared__ float se[Nq][Nq + 1];
    int b = blockIdx.x;
    for (int t = threadIdx.x; t < Nq * Nq; t += 64)
        se[t / Nq][t % Nq] = e[(long long)b * (Nq * Nq) + t];
    __syncthreads();

    int t = threadIdx.x;
    if (t < Nq) {                       // row softmax, i = t
        float mx = -1e30f;
        for (int j = 0; j < Nq; ++j) mx = fmaxf(mx, se[t][j]);
        float s = 0.f;
        for (int j = 0; j < Nq; ++j) s += expf(se[t][j] - mx);
        float inv = 1.f / s;
        for (int j = 0; j < Nq; ++j)
            arow[(long long)b * (Nq * Nq) + t * Nq + j] = expf(se[t][j] - mx) * inv;
    } else {                            // column softmax, j = t-32
        int jj = t - Nq;
        float mx = -1e30f;
        for (int i = 0; i < Nq; ++i) mx = fmaxf(mx, se[i][jj]);
        float s = 0.f;
        for (int i = 0; i < Nq; ++i) s += expf(se[i][jj] - mx);
        float inv = 1.f / s;
        for (int i = 0; i < Nq; ++i)
            acol[(long long)b * (Nq * Nq) + i * Nq + jj] = expf(se[i][jj] - mx) * inv;
    }
}

// ---------------------------------------------------------------------------
// row_ctx[i,:] = sum_j arow[i,j]*g[i,j,:] ;  col_ctx[j,:] = sum_i acol[i,j]*g[i,j,:]
// grid(N, 2, B), block 256.
// ---------------------------------------------------------------------------
__global__ __launch_bounds__(256)
void k_axisctx(const float* __restrict__ arow, const float* __restrict__ acol,
               const float* __restrict__ g, float* __restrict__ rowctx,
               float* __restrict__ colctx)
{
    int idx = blockIdx.x, mode = blockIdx.y, b = blockIdx.z;
    for (int w = threadIdx.x; w < WDq; w += 256) {
        float s = 0.f;
        if (mode == 0) {
            for (int j = 0; j < Nq; ++j)
                s += arow[(long long)b * (Nq * Nq) + idx * Nq + j] *
                     g[((long long)b * (Nq * Nq) + idx * Nq + j) * WDq + w];
            rowctx[((long long)(b * Nq + idx)) * WDq + w] = s;
        } else {
            for (int i = 0; i < Nq; ++i)
                s += acol[(long long)b * (Nq * Nq) + i * Nq + idx] *
                     g[((long long)b * (Nq * Nq) + i * Nq + idx) * WDq + w];
            colctx[((long long)(b * Nq + idx)) * WDq + w] = s;
        }
    }
}

// ---------------------------------------------------------------------------
// out[b,i,j,c] = gW[b,p,c] + rowW[b,i,c] + colW[b,j,c] + clf_b[c]
// grid(N*N, B), block 128.
// ---------------------------------------------------------------------------
__global__ __launch_bounds__(128)
void k_combine(const float* __restrict__ gW, const float* __restrict__ rowW,
               const float* __restrict__ colW, const float* __restrict__ clfb,
               float* __restrict__ out)
{
    int b = blockIdx.y, p = blockIdx.x;
    int i = p >> 5, j = p & 31;
    for (int c = threadIdx.x; c < Cq; c += 128) {
        out[((long long)b * (Nq * Nq) + p) * Cq + c] =
            gW[((long long)b * (Nq * Nq) + p) * Cq + c] +
            rowW[((long long)(b * Nq + i)) * Cq + c] +
            colW[((long long)(b * Nq + j)) * Cq + c] + clfb[c];
    }
}

// ---------------------------------------------------------------------------
// Host side
// ---------------------------------------------------------------------------
static void launch_gemm(const float* A, const float* Bm, float* C,
                        int M, int N, int K, int lda, int ldb, int ldc,
                        long long sA, long long sB, long long sC, int batch,
                        const float* bias, float biasScale, int doTanh,
                        hipStream_t stream)
{
    dim3 grid((N + BN - 1) / BN, (M + BM - 1) / BM, batch);
    gemm_f32_wmma<<<grid, 256, 0, stream>>>(A, Bm, C, M, N, K, lda, ldb, ldc,
                                            sA, sB, sC, bias, biasScale, doTanh);
}

// Workspace layout (float offsets). Total = 17,879,296 floats ~= 68.2 MiB.
#define OFF_REP     0LL
#define OFF_REPWS   98304LL
#define OFF_REPWO   196608LL
#define OFF_GA      294912LL
#define OFF_PA      1867776LL
#define OFF_SEQWC   6062080LL
#define OFF_CTX     9207808LL
#define OFF_PVEC    12353536LL
#define OFF_G       12943360LL
#define OFF_ABLT    16089088LL
#define OFF_TANHT   16199680LL
#define OFF_E       17248256LL
#define OFF_AROW    17252352LL
#define OFF_ACOL    17256448LL
#define OFF_ROWCTX  17260544LL
#define OFF_COLCTX  17358848LL
#define OFF_ROWW    17457152LL
#define OFF_COLW    17469568LL
#define OFF_GW      17481984LL

extern "C" void kernel_launch(void* const* d_in, const int* in_sizes, int n_in,
                              void* d_out, int out_size, void* d_ws, size_t ws_size,
                              hipStream_t stream)
{
    const float* seq   = (const float*)d_in[0];   // [B,S,H]
    const float* att   = (const float*)d_in[1];   // [B,Hh,S,S]
    const int*   ep    = (const int*)  d_in[2];   // [B,N,M]
    const float* W_s   = (const float*)d_in[3];   // [H,Wd]
    const float* W_o   = (const float*)d_in[4];   // [H,Wd]
    const float* W_c   = (const float*)d_in[5];   // [H,Wd]
    const float* A_bl  = (const float*)d_in[6];   // [Wd,d,d]
    const float* b_bl  = (const float*)d_in[7];   // [Wd]
    const float* W_at  = (const float*)d_in[8];   // [Wd,AD]
    const float* v_at  = (const float*)d_in[9];   // [AD]
    const float* clf_W = (const float*)d_in[10];  // [Wd,C]
    const float* clf_b = (const float*)d_in[11];  // [C]
    float* out = (float*)d_out;                   // [B,N,N,C]

    float* ws = (float*)d_ws;
    float* rep    = ws + OFF_REP;
    float* repWs  = ws + OFF_REPWS;
    float* repWo  = ws + OFF_REPWO;
    float* ga     = ws + OFF_GA;
    float* pa     = ws + OFF_PA;
    float* seqWc  = ws + OFF_SEQWC;
    float* ctx    = ws + OFF_CTX;
    float* Pvec   = ws + OFF_PVEC;
    float* g      = ws + OFF_G;
    float* AblT   = ws + OFF_ABLT;
    float* tanhT  = ws + OFF_TANHT;
    float* e      = ws + OFF_E;
    float* arow   = ws + OFF_AROW;
    float* acol   = ws + OFF_ACOL;
    float* rowctx = ws + OFF_ROWCTX;
    float* colctx = ws + OFF_COLCTX;
    float* rowW   = ws + OFF_ROWW;
    float* colW   = ws + OFF_COLW;
    float* gW     = ws + OFF_GW;

    // 1) entity reps (logsumexp pooling) + entity attention rows
    k_rep<<<dim3(Nq, Bq), 256, 0, stream>>>(seq, ep, rep);
    k_ga<<<dim3(HHq, Nq, Bq), 256, 0, stream>>>(att, ep, ga);

    // 2) rep projections: [B*N,768] @ [768,768]  (WMMA + async staging)
    launch_gemm(rep, W_s, repWs, Bq * Nq, WDq, Hq, Hq, WDq, WDq,
                0, 0, 0, 1, nullptr, 0.f, 0, stream);
    launch_gemm(rep, W_o, repWo, Bq * Nq, WDq, Hq, Hq, WDq, WDq,
                0, 0, 0, 1, nullptr, 0.f, 0, stream);

    // 3) seqWc = seq @ W_c : [B*S,768] @ [768,768]
    launch_gemm(seq, W_c, seqWc, Bq * Sq, WDq, Hq, Hq, WDq, WDq,
                0, 0, 0, 1, nullptr, 0.f, 0, stream);

    // 4) pairwise localized attention, normalized over S
    k_pa<<<dim3(Nq * Nq, Bq), 256, 0, stream>>>(ga, pa);

    // 5) ctx = pa @ seqWc : batched [1024,1024] @ [1024,768]
    launch_gemm(pa, seqWc, ctx, Nq * Nq, WDq, Sq, Sq, WDq, WDq,
                (long long)(Nq * Nq) * Sq, (long long)Sq * WDq,
                (long long)(Nq * Nq) * WDq, Bq, nullptr, 0.f, 0, stream);

    // 6) tanh + per-pair grouped outer products -> Pvec [B*1024,144]
    k_zsP<<<dim3(Nq * Nq, Bq), 256, 0, stream>>>(ctx, repWs, repWo, Pvec);

    // 7) g = Pvec @ A_bl^T + SPLIT_K * b_bl : [4096,144] @ [144,768]
    k_transAbl<<<(DDq * DDq * WDq + 255) / 256, 256, 0, stream>>>(A_bl, AblT);
    launch_gemm(Pvec, AblT, g, Bq * Nq * Nq, WDq, DDq * DDq,
                DDq * DDq, WDq, WDq, 0, 0, 0, 1, b_bl, (float)SKq, 0, stream);

    // 8) tanhT = tanh(g @ W_attn) : [4096,768] @ [768,256], fused tanh
    launch_gemm(g, W_at, tanhT, Bq * Nq * Nq, ADq, WDq, WDq, ADq, ADq,
                0, 0, 0, 1, nullptr, 0.f, 1, stream);

    // 9) e = tanhT @ v_attn ; axis softmaxes ; axis contexts
    k_e<<<dim3(Nq * Nq, Bq), 256, 0, stream>>>(tanhT, v_at, e);
    k_softmax_axes<<<Bq, 64, 0, stream>>>(e, arow, acol);
    k_axisctx<<<dim3(Nq, 2, Bq), 256, 0, stream>>>(arow, acol, g, rowctx, colctx);

    // 10) classifier, distributed over the broadcast sum:
    //     out = g@clf_W + (rowctx@clf_W)[i] + (colctx@clf_W)[j] + clf_b
    launch_gemm(rowctx, clf_W, rowW, Bq * Nq, Cq, WDq, WDq, Cq, Cq,
                0, 0, 0, 1, nullptr, 0.f, 0, stream);
    launch_gemm(colctx, clf_W, colW, Bq * Nq, Cq, WDq, WDq, Cq, Cq,
                0, 0, 0, 1, nullptr, 0.f, 0, stream);
    launch_gemm(g, clf_W, gW, Bq * Nq * Nq, Cq, WDq, WDq, Cq, Cq,
                0, 0, 0, 1, nullptr, 0.f, 0, stream);

    k_combine<<<dim3(Nq * Nq, Bq), 128, 0, stream>>>(gW, rowW, colW, clf_b, out);
}